// MultiSemanticHyperConvNetwork_23742579212952
// MI455X (gfx1250) — compile-verified
//
#include <hip/hip_runtime.h>
#include <hip/hip_bf16.h>

typedef __attribute__((ext_vector_type(2))) float v2f;
typedef __attribute__((ext_vector_type(8))) float v8f;

#define E_DIM 64

// ---------------------------------------------------------------------------
// Zero a float buffer (float4-wide).
// ---------------------------------------------------------------------------
__global__ void zero_f32_kernel(float* __restrict__ p, long n_float4) {
    long t = (long)blockIdx.x * blockDim.x + threadIdx.x;
    if (t >= n_float4) return;
    ((float4*)p)[t] = make_float4(0.f, 0.f, 0.f, 0.f);
}

// ---------------------------------------------------------------------------
// Stage 1: ga/sa/pa[u] += val * {geo,seq,col}[poi]   (COO, 16 threads/edge,
// each thread owns a float4 slice of the E=64 row).
// ---------------------------------------------------------------------------
__global__ void spmm_up_kernel(const int* __restrict__ rows,
                               const int* __restrict__ cols,
                               const float* __restrict__ vals,
                               const float* __restrict__ geo,
                               const float* __restrict__ seq,
                               const float* __restrict__ colp,
                               float* __restrict__ ga,
                               float* __restrict__ sa,
                               float* __restrict__ pa,
                               long total) {
    long t = (long)blockIdx.x * blockDim.x + threadIdx.x;
    if (t >= total) return;
    const long edge = t >> 4;
    const int  c    = (int)(t & 15) * 4;
    const int  u    = rows[edge];
    const int  poi  = cols[edge];
    const float v   = vals[edge];
    const long pi = (long)poi * E_DIM + c;
    const long ui = (long)u   * E_DIM + c;
    const float4 g = *(const float4*)(geo  + pi);
    const float4 s = *(const float4*)(seq  + pi);
    const float4 p = *(const float4*)(colp + pi);
    atomicAdd(ga + ui + 0, v * g.x); atomicAdd(ga + ui + 1, v * g.y);
    atomicAdd(ga + ui + 2, v * g.z); atomicAdd(ga + ui + 3, v * g.w);
    atomicAdd(sa + ui + 0, v * s.x); atomicAdd(sa + ui + 1, v * s.y);
    atomicAdd(sa + ui + 2, v * s.z); atomicAdd(sa + ui + 3, v * s.w);
    atomicAdd(pa + ui + 0, v * p.x); atomicAdd(pa + ui + 1, v * p.y);
    atomicAdd(pa + ui + 2, v * p.z); atomicAdd(pa + ui + 3, v * p.w);
}

// ---------------------------------------------------------------------------
// Stage 2: fusion GEMM with V_WMMA_F32_16X16X4_F32.
// One wave32 computes a 16(user) x 64(e) tile of
//   msg_emb[u,e] = sum_f msg[u,f] * W[e,f],  f in [0,448)
// msg feature blocks b=0..6 are {g, s, p, g*s, g*p, s*p, g*s*p} built on the
// fly from ga/sa/pa. Then hg = me + users + me*users with me = msg_emb + bias.
//
// A-operand (16x4 f32): lanes 0-15 hold M=lane, K={0,1} in vgpr{0,1};
// lanes 16-31 hold K={2,3}. B-operand (4x16) mirrors: lane%16 = N column,
// lane/16 selects K pair. C/D (16x16 f32, 8 vgprs): lane%16 = N,
// M = (lane/16)*8 + vgpr_index.
// ---------------------------------------------------------------------------
__global__ void fusion_wmma_kernel(const float* __restrict__ ga,
                                   const float* __restrict__ sa,
                                   const float* __restrict__ pa,
                                   const float* __restrict__ W,     // [64,448]
                                   const float* __restrict__ bias,  // [64]
                                   const float* __restrict__ users, // [U,64]
                                   float* __restrict__ hg,          // [U,64]
                                   int n_users) {
    const int wave  = blockIdx.x * (blockDim.x >> 5) + (threadIdx.x >> 5);
    const int lane  = threadIdx.x & 31;
    const int ubase = wave * 16;
    if (ubase >= n_users) return;   // wave-uniform: EXEC stays all-ones

    const int r     = lane & 15;    // A row / B col / D col (lane group)
    const int khalf = lane >> 4;    // K pair select
    const long urow = (long)(ubase + r) * E_DIM;

    v8f acc0 = {}, acc1 = {}, acc2 = {}, acc3 = {};

#define FUSE_WMMA_N(ACC, NT)                                                   \
    do {                                                                       \
        const float2 w2 = *(const float2*)(W + (long)((NT)*16 + r) * 448 +     \
                                           kk + khalf * 2);                    \
        v2f bv; bv[0] = w2.x; bv[1] = w2.y;                                    \
        ACC = __builtin_amdgcn_wmma_f32_16x16x4_f32(                           \
            false, av, false, bv, (short)0, ACC, false, false);                \
    } while (0)

#pragma unroll
    for (int b = 0; b < 7; ++b) {
        for (int e0 = 0; e0 < E_DIM; e0 += 4) {
            const int kk = b * E_DIM + e0;          // feature index base
            const int e  = e0 + khalf * 2;          // this lane's e pair
            v2f av;
            if (b == 0) {
                const float2 g = *(const float2*)(ga + urow + e);
                av[0] = g.x; av[1] = g.y;
            } else if (b == 1) {
                const float2 s = *(const float2*)(sa + urow + e);
                av[0] = s.x; av[1] = s.y;
            } else if (b == 2) {
                const float2 p = *(const float2*)(pa + urow + e);
                av[0] = p.x; av[1] = p.y;
            } else if (b == 3) {
                const float2 g = *(const float2*)(ga + urow + e);
                const float2 s = *(const float2*)(sa + urow + e);
                av[0] = g.x * s.x; av[1] = g.y * s.y;
            } else if (b == 4) {
                const float2 g = *(const float2*)(ga + urow + e);
                const float2 p = *(const float2*)(pa + urow + e);
                av[0] = g.x * p.x; av[1] = g.y * p.y;
            } else if (b == 5) {
                const float2 s = *(const float2*)(sa + urow + e);
                const float2 p = *(const float2*)(pa + urow + e);
                av[0] = s.x * p.x; av[1] = s.y * p.y;
            } else {
                const float2 g = *(const float2*)(ga + urow + e);
                const float2 s = *(const float2*)(sa + urow + e);
                const float2 p = *(const float2*)(pa + urow + e);
                av[0] = g.x * s.x * p.x; av[1] = g.y * s.y * p.y;
            }
            FUSE_WMMA_N(acc0, 0);
            FUSE_WMMA_N(acc1, 1);
            FUSE_WMMA_N(acc2, 2);
            FUSE_WMMA_N(acc3, 3);
        }
    }
#undef FUSE_WMMA_N

    // Epilogue: me = acc + bias; hg = me + users + me*users
#define FUSE_EPI(ACC, NT)                                                      \
    do {                                                                       \
        const int col = (NT) * 16 + r;                                         \
        const float bf = bias[col];                                            \
        _Pragma("unroll")                                                      \
        for (int j = 0; j < 8; ++j) {                                          \
            const int m = khalf * 8 + j;                                       \
            const long idx = (long)(ubase + m) * E_DIM + col;                  \
            const float me = ACC[j] + bf;                                      \
            const float us = users[idx];                                       \
            hg[idx] = me + us + me * us;                                       \
        }                                                                      \
    } while (0)

    FUSE_EPI(acc0, 0);
    FUSE_EPI(acc1, 1);
    FUSE_EPI(acc2, 2);
    FUSE_EPI(acc3, 3);
#undef FUSE_EPI
}

// ---------------------------------------------------------------------------
// Stage 3: out[poi] += (num_layers/2) * pu_val * hg[u]  (scatter; out was
// pre-initialized with init_pois_embs). 16 threads/edge, float4 slices.
// ---------------------------------------------------------------------------
__global__ void spmm_pu_kernel(const int* __restrict__ up_rows,
                               const int* __restrict__ up_cols,
                               const float* __restrict__ pu_vals,
                               const float* __restrict__ hg,
                               const int* __restrict__ num_layers,
                               float* __restrict__ out,
                               long total) {
    long t = (long)blockIdx.x * blockDim.x + threadIdx.x;
    if (t >= total) return;
    const float scale = 0.5f * (float)num_layers[0];
    const long edge = t >> 4;
    const int  c    = (int)(t & 15) * 4;
    const int  u    = up_rows[edge];   // gather index into hg
    const int  poi  = up_cols[edge];   // scatter index into out
    const float v   = pu_vals[edge] * scale;
    const float4 h = *(const float4*)(hg + (long)u * E_DIM + c);
    float* o = out + (long)poi * E_DIM + c;
    atomicAdd(o + 0, v * h.x);
    atomicAdd(o + 1, v * h.y);
    atomicAdd(o + 2, v * h.z);
    atomicAdd(o + 3, v * h.w);
}

// ---------------------------------------------------------------------------
extern "C" void kernel_launch(void* const* d_in, const int* in_sizes, int n_in,
                              void* d_out, int out_size, void* d_ws, size_t ws_size,
                              hipStream_t stream) {
    const float* init_p  = (const float*)d_in[0];
    const float* col_p   = (const float*)d_in[1];
    const float* geo_p   = (const float*)d_in[2];
    const float* seq_p   = (const float*)d_in[3];
    const float* users   = (const float*)d_in[4];
    const int*   up_rows = (const int*)d_in[5];
    const int*   up_cols = (const int*)d_in[6];
    const float* up_vals = (const float*)d_in[7];
    const float* pu_vals = (const float*)d_in[8];
    const float* W       = (const float*)d_in[9];
    const float* bias    = (const float*)d_in[10];
    const int*   n_lay   = (const int*)d_in[11];

    const int U   = in_sizes[4] / E_DIM;
    const int P   = in_sizes[0] / E_DIM;
    const long NNZ = in_sizes[5];

    float* out = (float*)d_out;

    float* ga = (float*)d_ws;
    float* sa = ga + (size_t)U * E_DIM;
    float* pa = sa + (size_t)U * E_DIM;
    float* hg = pa + (size_t)U * E_DIM;

    // Zero the three user aggregates.
    {
        long nf4 = (long)U * E_DIM * 3 / 4;
        long blocks = (nf4 + 255) / 256;
        zero_f32_kernel<<<(int)blocks, 256, 0, stream>>>(ga, nf4);
    }

    // out = init_pois_embs (then spmm_pu accumulates propag on top).
    (void)hipMemcpyAsync(out, init_p, (size_t)P * E_DIM * sizeof(float),
                         hipMemcpyDeviceToDevice, stream);

    // Stage 1: three COO SpMMs in one pass.
    {
        long total = NNZ * 16;
        long blocks = (total + 255) / 256;
        spmm_up_kernel<<<(int)blocks, 256, 0, stream>>>(
            up_rows, up_cols, up_vals, geo_p, seq_p, col_p, ga, sa, pa, total);
    }

    // Stage 2: WMMA fusion GEMM + gated residual -> hg.
    {
        int tiles  = (U + 15) / 16;        // 16-user tiles, one wave each
        int blocks = (tiles + 3) / 4;      // 4 waves (128 threads) per block
        fusion_wmma_kernel<<<blocks, 128, 0, stream>>>(
            ga, sa, pa, W, bias, users, hg, U);
    }

    // Stage 3: transpose-structure SpMM into out, scaled by num_layers/2.
    {
        long total = NNZ * 16;
        long blocks = (total + 255) / 256;
        spmm_pu_kernel<<<(int)blocks, 256, 0, stream>>>(
            up_rows, up_cols, pu_vals, hg, n_lay, out, total);
    }
}